// TransitionDown_8650064134300
// MI455X (gfx1250) — compile-verified
//
#include <hip/hip_runtime.h>
#include <stdint.h>

#define N_PTS   16384
#define M_CTR   4096
#define NSAMP   16
#define CIN     64
#define COUT    128
#define KDIM    67      // 3 + CIN (W LDS row stride; 67 mod 64 = 3 -> conflict-free)
#define KPAD    68      // A-tile K padded to multiple of 4 (pad col is zero)
#define EPSI    1e-5f
#define GEMM_BLOCKS (M_CTR / 8)   // 8 waves (centers) per 256-thread block

typedef float v2f __attribute__((ext_vector_type(2)));
typedef float v8f __attribute__((ext_vector_type(8)));
typedef unsigned int u32x4 __attribute__((ext_vector_type(4)));
typedef int i32x4 __attribute__((ext_vector_type(4)));
typedef int i32x8 __attribute__((ext_vector_type(8)));

#if defined(__gfx1250__) && __has_builtin(__builtin_amdgcn_tensor_load_to_lds) && \
    __has_builtin(__builtin_amdgcn_s_wait_tensorcnt)
#define HAVE_TDM 1
#else
#define HAVE_TDM 0
#endif

__device__ __forceinline__ unsigned long long shflx64(unsigned long long v, int m) {
    unsigned lo = (unsigned)v, hi = (unsigned)(v >> 32);
    lo = __shfl_xor(lo, m, 32);
    hi = __shfl_xor(hi, m, 32);
    return ((unsigned long long)hi << 32) | lo;
}

// ---------------------------------------------------------------------------
// Kernel 1: furthest point sampling. One persistent 1024-thread workgroup.
// Packed key = (dist_bits << 32) | (0xFFFFFFFF - idx): max-reduce picks the
// largest distance, ties resolve to the smallest index (jnp.argmax semantics).
// Writes n_p (out[0:3M)) and n_n (out[3M:6M)).
// ---------------------------------------------------------------------------
__global__ __launch_bounds__(1024) void fps_kernel(const float* __restrict__ p,
                                                   const float* __restrict__ n,
                                                   float* __restrict__ out) {
    __shared__ int sIdx[M_CTR];
    __shared__ unsigned long long warpRed[32];
    __shared__ int sWinner;

    const int tid  = threadIdx.x;
    const int lane = tid & 31;
    const int wave = tid >> 5;

    float dd[16];
#pragma unroll
    for (int j = 0; j < 16; ++j) dd[j] = 1e10f;

    if (tid == 0) sIdx[0] = 0;
    float cx = p[0], cy = p[1], cz = p[2];

    for (int it = 1; it < M_CTR; ++it) {
        unsigned long long best = 0ull;
#pragma unroll
        for (int j = 0; j < 16; ++j) {
            const int i = j * 1024 + tid;
            float dx = p[i * 3 + 0] - cx;
            float dy = p[i * 3 + 1] - cy;
            float dz = p[i * 3 + 2] - cz;
            float d  = dx * dx + dy * dy + dz * dz;
            d = fminf(d, dd[j]);
            dd[j] = d;
            unsigned long long key =
                ((unsigned long long)__float_as_uint(d) << 32) |
                (unsigned)(0xFFFFFFFFu - (unsigned)i);
            best = (key > best) ? key : best;
        }
#pragma unroll
        for (int s = 16; s >= 1; s >>= 1) {
            unsigned long long o = shflx64(best, s);
            best = (o > best) ? o : best;
        }
        if (lane == 0) warpRed[wave] = best;
        __syncthreads();
        if (wave == 0) {
            unsigned long long b = warpRed[lane];
#pragma unroll
            for (int s = 16; s >= 1; s >>= 1) {
                unsigned long long o = shflx64(b, s);
                b = (o > b) ? o : b;
            }
            if (lane == 0) {
                int widx = (int)(0xFFFFFFFFu - (unsigned)(b & 0xFFFFFFFFull));
                sIdx[it] = widx;
                sWinner  = widx;
            }
        }
        __syncthreads();
        const int w = sWinner;
        cx = p[w * 3 + 0];
        cy = p[w * 3 + 1];
        cz = p[w * 3 + 2];
    }
    __syncthreads();

    // gather n_p / n_n
    for (int i = tid; i < M_CTR; i += 1024) {
        const int j = sIdx[i];
        out[i * 3 + 0] = p[j * 3 + 0];
        out[i * 3 + 1] = p[j * 3 + 1];
        out[i * 3 + 2] = p[j * 3 + 2];
        out[M_CTR * 3 + i * 3 + 0] = n[j * 3 + 0];
        out[M_CTR * 3 + i * 3 + 1] = n[j * 3 + 1];
        out[M_CTR * 3 + i * 3 + 2] = n[j * 3 + 2];
    }
}

// ---------------------------------------------------------------------------
// Kernel 2: exact kNN (16-NN), one wave per center. Per-lane register top-16
// via replace-max over a 512-point stripe, then a 16-round wave-min merge on
// 64-bit keys (dist_bits << 32 | point_idx): ascending distances, ties to the
// smaller index — matches lax.top_k ordering.
// ---------------------------------------------------------------------------
__global__ __launch_bounds__(256) void knn_kernel(const float* __restrict__ p,
                                                  const float* __restrict__ n_p,
                                                  int* __restrict__ knn) {
    const int lane = threadIdx.x & 31;
    const int wave = threadIdx.x >> 5;
    const int c    = blockIdx.x * 8 + wave;

    const float cx = n_p[c * 3 + 0];
    const float cy = n_p[c * 3 + 1];
    const float cz = n_p[c * 3 + 2];

    float bd[16];
    int   bi[16];
#pragma unroll
    for (int j = 0; j < 16; ++j) { bd[j] = 3.0e38f; bi[j] = 0x7FFFFFFF; }
    float worst = 3.0e38f;
    int   wslot = 0;

    for (int i = lane; i < N_PTS; i += 32) {
        if (i + 512 < N_PTS) __builtin_prefetch(&p[(i + 512) * 3], 0, 0);
        float dx = p[i * 3 + 0] - cx;
        float dy = p[i * 3 + 1] - cy;
        float dz = p[i * 3 + 2] - cz;
        float d  = dx * dx + dy * dy + dz * dz;
        if (d < worst) {
#pragma unroll
            for (int j = 0; j < 16; ++j)
                if (j == wslot) { bd[j] = d; bi[j] = i; }
            worst = -1.0f;
#pragma unroll
            for (int j = 0; j < 16; ++j)
                if (bd[j] > worst) { worst = bd[j]; wslot = j; }
        }
    }

    unsigned mask = 0u;
    for (int r = 0; r < 16; ++r) {
        unsigned long long k = ~0ull;
        int lslot = 0;
#pragma unroll
        for (int j = 0; j < 16; ++j) {
            unsigned long long kk =
                ((unsigned long long)__float_as_uint(bd[j]) << 32) | (unsigned)bi[j];
            if (!(mask & (1u << j)) && kk < k) { k = kk; lslot = j; }
        }
        unsigned long long kmin = k;
#pragma unroll
        for (int s = 16; s >= 1; s >>= 1) {
            unsigned long long o = shflx64(kmin, s);
            kmin = (o < kmin) ? o : kmin;
        }
        if (kmin == k) mask |= (1u << lslot);     // unique keys -> one winner
        if (lane == 0) knn[c * NSAMP + r] = (int)(kmin & 0xFFFFFFFFull);
    }
}

// ---------------------------------------------------------------------------
// TDM: DMA W[128*67] fp32 (contiguous 34KB) global -> LDS as a 1-D tile.
// D# per ISA §8: group0 = {count=1, lds_addr, global_addr[31:0],
// addr[56:32]|type=2<<30}; group1 = data_size=4B, tensor/tile dim0 = 8576.
// ---------------------------------------------------------------------------
#if HAVE_TDM
__device__ __forceinline__ void tdm_load_w(float* ldsDst, const float* __restrict__ gsrc) {
    const unsigned long long ga = (unsigned long long)(uintptr_t)gsrc;
    const unsigned lds   = (unsigned)(uintptr_t)ldsDst;   // low 32 bits = LDS offset
    const unsigned nelem = COUT * KDIM;                   // 8576 dwords
    u32x4 g0 = { 0x1u,                                    // count=1, no gather
                 lds,                                     // lds_addr
                 (unsigned)ga,                            // global_addr[31:0]
                 (unsigned)((ga >> 32) & 0x1FFFFFFu) | (2u << 30) };  // [56:32] | type=2
    i32x8 g1 = { (int)(2u << 16),                         // data_size = 4 bytes
                 (int)((nelem & 0xFFFFu) << 16),          // tensor_dim0[15:0]  (bits 63:48)
                 (int)(((nelem >> 16) & 0xFFFFu) | (1u << 16)), // tensor_dim0[31:16], tensor_dim1=1
                 (int)((nelem & 0xFFFFu) << 16),          // tile_dim0 (bits 127:112)
                 0,                                       // tile_dim1=0 (unused), tile_dim2=0
                 (int)nelem,                              // tensor_dim0_stride[31:0]
                 0, 0 };
    i32x4 g2 = { 0, 0, 0, 0 };
    i32x4 g3 = { 0, 0, 0, 0 };
#if __clang_major__ >= 23
    i32x8 g4 = { 0, 0, 0, 0, 0, 0, 0, 0 };
    __builtin_amdgcn_tensor_load_to_lds(g0, g1, g2, g3, g4, 0);
#else
    __builtin_amdgcn_tensor_load_to_lds(g0, g1, g2, g3, 0);
#endif
}
#endif

// ---------------------------------------------------------------------------
// Shared GEMM core. W lives in LDS at stride KDIM=67 (exactly the global
// layout so TDM/plain copy is contiguous); A-tile[16][68] with K=67 zeroed so
// the one-past-the-end B element (index COUT*KDIM, zeroed) contributes 0.
// Fully-unrolled 17 x 8 V_WMMA_F32_16X16X4_F32 chain.
// ---------------------------------------------------------------------------
__device__ __forceinline__ void stage_and_mma(
    const float* __restrict__ p, const float* __restrict__ x,
    const float* __restrict__ n_p, const int* __restrict__ knn,
    const float* __restrict__ Wg,
    float* Wl /* [COUT*KDIM + 1] */, float (&Apad)[NSAMP][KPAD],
    int m, int tid, int lane, int wave, v8f (&acc)[8]) {
#if HAVE_TDM
    if (wave == 0) tdm_load_w(Wl, Wg);          // async DMA, TENSORcnt tracked
#else
    for (int e = tid; e < COUT * KDIM; e += 256) Wl[e] = Wg[e];
#endif
    if (tid == 0) Wl[COUT * KDIM] = 0.0f;       // tail dword read at (row127,k=67)

    // stage this wave's A tile: rows = 16 neighbors of center m
    for (int e = lane; e < NSAMP * KPAD; e += 32) {
        const int s = e / KPAD, k = e - s * KPAD;
        const int j = knn[m * NSAMP + s];
        float v;
        if (k < 3)          v = p[j * 3 + k] - n_p[m * 3 + k];
        else if (k < KDIM)  v = x[j * CIN + (k - 3)];
        else                v = 0.0f;
        Apad[s][k] = v;
    }
#if HAVE_TDM
    if (wave == 0) __builtin_amdgcn_s_wait_tensorcnt(0);
#endif
    __syncthreads();

    const int r  = lane & 15;
    const int hi = lane >> 4;
#pragma unroll
    for (int k0 = 0; k0 < KPAD; k0 += 4) {
        v2f a;
        a.x = Apad[r][k0 + 2 * hi];
        a.y = Apad[r][k0 + 2 * hi + 1];
#pragma unroll
        for (int ct = 0; ct < 8; ++ct) {
            v2f b;
            b.x = Wl[(ct * 16 + r) * KDIM + k0 + 2 * hi];
            b.y = Wl[(ct * 16 + r) * KDIM + k0 + 2 * hi + 1];
            acc[ct] = __builtin_amdgcn_wmma_f32_16x16x4_f32(
                false, a, false, b, (short)0, acc[ct], false, false);
        }
    }
}

// ---------------------------------------------------------------------------
// Kernel 3: GEMM pass 1 — deterministic per-block channel sum / sum-of-squares
// (wave shfl tree -> LDS -> fixed-order 8-way sum), written per block.
// ---------------------------------------------------------------------------
__global__ __launch_bounds__(256) void gemm_stats_kernel(
    const float* __restrict__ p, const float* __restrict__ x,
    const float* __restrict__ n_p, const float* __restrict__ Wg,
    const int* __restrict__ knn,
    float* __restrict__ blkSum, float* __restrict__ blkSq) {
    __shared__ float Wl[COUT * KDIM + 1];
    __shared__ float Apad[8][NSAMP][KPAD];
    __shared__ float wSum[8][COUT];
    __shared__ float wSq[8][COUT];

    const int tid  = threadIdx.x;
    const int lane = tid & 31;
    const int wave = tid >> 5;
    const int m    = blockIdx.x * 8 + wave;

    v8f acc[8] = {};
    stage_and_mma(p, x, n_p, knn, Wg, Wl, Apad[wave], m, tid, lane, wave, acc);

    const int r = lane & 15;
#pragma unroll
    for (int ct = 0; ct < 8; ++ct) {
        float s = 0.0f, q = 0.0f;
#pragma unroll
        for (int j = 0; j < 8; ++j) {
            const float v = acc[ct][j];
            s += v;
            q += v * v;
        }
        s += __shfl_xor(s, 16, 32);
        q += __shfl_xor(q, 16, 32);
        if (lane < 16) {
            wSum[wave][ct * 16 + r] = s;
            wSq[wave][ct * 16 + r]  = q;
        }
    }
    __syncthreads();
    if (tid < COUT) {
        float s = 0.0f, q = 0.0f;
#pragma unroll
        for (int w = 0; w < 8; ++w) { s += wSum[w][tid]; q += wSq[w][tid]; }
        blkSum[blockIdx.x * COUT + tid] = s;
        blkSq[blockIdx.x * COUT + tid]  = q;
    }
}

// ---------------------------------------------------------------------------
// Kernel 4: finalize BN stats -> scale/bias; also writes n_o (last element).
// ---------------------------------------------------------------------------
__global__ __launch_bounds__(256) void stats_kernel(
    const float* __restrict__ blkSum, const float* __restrict__ blkSq,
    const float* __restrict__ gamma, const float* __restrict__ beta,
    float* __restrict__ scaleBias, float* __restrict__ out) {
    __shared__ float sSum[COUT];
    __shared__ float sSq[COUT];
    const int tid = threadIdx.x;
    if (tid < COUT) {
        float s = 0.0f;
        for (int b = 0; b < GEMM_BLOCKS; ++b) s += blkSum[b * COUT + tid];
        sSum[tid] = s;
    } else {
        const int ch = tid - COUT;
        float q = 0.0f;
        for (int b = 0; b < GEMM_BLOCKS; ++b) q += blkSq[b * COUT + ch];
        sSq[ch] = q;
    }
    __syncthreads();
    if (tid < COUT) {
        const float cnt  = (float)(M_CTR * NSAMP);
        const float mean = sSum[tid] / cnt;
        const float var  = sSq[tid] / cnt - mean * mean;
        const float sc   = gamma[tid] * rsqrtf(var + EPSI);
        scaleBias[tid]        = sc;
        scaleBias[COUT + tid] = beta[tid] - mean * sc;
    }
    if (tid == 0) out[M_CTR * 6 + M_CTR * COUT] = (float)M_CTR;  // n_o
}

// ---------------------------------------------------------------------------
// Kernel 5: GEMM pass 2 — recompute tiles, affine+ReLU, max over 16 neighbors
// (max over tile rows: 8 in-lane accs + one shfl_xor(16)).
// ---------------------------------------------------------------------------
__global__ __launch_bounds__(256) void gemm_out_kernel(
    const float* __restrict__ p, const float* __restrict__ x,
    const float* __restrict__ n_p, const float* __restrict__ Wg,
    const int* __restrict__ knn, const float* __restrict__ scaleBias,
    float* __restrict__ outY) {
    __shared__ float Wl[COUT * KDIM + 1];
    __shared__ float Apad[8][NSAMP][KPAD];

    const int tid  = threadIdx.x;
    const int lane = tid & 31;
    const int wave = tid >> 5;
    const int m    = blockIdx.x * 8 + wave;

    v8f acc[8] = {};
    stage_and_mma(p, x, n_p, knn, Wg, Wl, Apad[wave], m, tid, lane, wave, acc);

    const int r = lane & 15;
#pragma unroll
    for (int ct = 0; ct < 8; ++ct) {
        const int col = ct * 16 + r;
        const float sc = scaleBias[col];
        const float bs = scaleBias[COUT + col];
        float mx = 0.0f;  // ReLU floor
#pragma unroll
        for (int j = 0; j < 8; ++j)
            mx = fmaxf(mx, fmaf(acc[ct][j], sc, bs));
        mx = fmaxf(mx, __shfl_xor(mx, 16, 32));
        if (lane < 16) outY[m * COUT + col] = mx;
    }
}

// ---------------------------------------------------------------------------
extern "C" void kernel_launch(void* const* d_in, const int* in_sizes, int n_in,
                              void* d_out, int out_size, void* d_ws, size_t ws_size,
                              hipStream_t stream) {
    (void)in_sizes; (void)n_in; (void)out_size; (void)ws_size;
    const float* p     = (const float*)d_in[0];
    const float* n     = (const float*)d_in[1];
    const float* x     = (const float*)d_in[2];
    /* d_in[3] = o (unused: single batch, sizes are static) */
    const float* W     = (const float*)d_in[4];
    const float* gamma = (const float*)d_in[5];
    const float* beta  = (const float*)d_in[6];
    float* out = (float*)d_out;

    char* ws = (char*)d_ws;
    int*   knn       = (int*)(ws);                                  // 256 KB
    float* blkSum    = (float*)(ws + (size_t)M_CTR * NSAMP * 4);
    float* blkSq     = (float*)(ws + (size_t)M_CTR * NSAMP * 4 + (size_t)GEMM_BLOCKS * COUT * 4);
    float* scaleBias = (float*)(ws + (size_t)M_CTR * NSAMP * 4 + (size_t)GEMM_BLOCKS * COUT * 8);

    fps_kernel<<<1, 1024, 0, stream>>>(p, n, out);
    knn_kernel<<<GEMM_BLOCKS, 256, 0, stream>>>(p, out, knn);
    gemm_stats_kernel<<<GEMM_BLOCKS, 256, 0, stream>>>(p, x, out, W, knn, blkSum, blkSq);
    stats_kernel<<<1, 256, 0, stream>>>(blkSum, blkSq, gamma, beta, scaleBias, out);
    gemm_out_kernel<<<GEMM_BLOCKS, 256, 0, stream>>>(p, x, out, W, knn, scaleBias,
                                                     out + (size_t)M_CTR * 6);
}